// CorVolume_85048942395513
// MI455X (gfx1250) — compile-verified
//
#include <hip/hip_runtime.h>
#include <hip/hip_bf16.h>

typedef __attribute__((ext_vector_type(2))) float v2f;
typedef __attribute__((ext_vector_type(8))) float v8f;

// Problem sizes (match reference setup_inputs)
constexpr int B_ = 8, C_ = 64, H_ = 160, W_ = 320, D_ = 48;
constexpr int WCHUNK  = 64;             // w columns per block
constexpr int NWCHUNK = W_ / WCHUNK;    // 5
constexpr int RW      = WCHUNK + D_;    // 112 right columns staged (band overhang)
constexpr int CP      = C_ / 2;         // 32 channel pairs
// LDS row strides in float2 units; need 2*stride % 64 == 32 so the two 16-lane
// halves of a wave (rows cp vs cp+1, same w-group) hit disjoint bank sets on b64.
constexpr int LS2 = 80;                 // left:  2*80  % 64 = 32
constexpr int RS2 = 112;                // right: 2*112 % 64 = 32

__global__ __launch_bounds__(128) void corr_volume_wmma(
    const float* __restrict__ left, const float* __restrict__ right,
    float* __restrict__ out)
{
    // Channel-pair interleaved: sX[cp*stride + w] = { src[2cp][w], src[2cp+1][w] }
    // => every WMMA A/B fragment is ONE contiguous ds_load_b64.
    __shared__ v2f sL[CP * LS2];   // 20,480 B
    __shared__ v2f sR[CP * RS2];   // 28,672 B

    const int g  = blockIdx.x;
    const int wc = g % NWCHUNK;
    const int h  = (g / NWCHUNK) % H_;
    const int b  = g / (NWCHUNK * H_);
    const int w0 = wc * WCHUNK;

    const int HW = H_ * W_;
    const float* lbase = left  + ((size_t)(b * C_) * H_ + h) * W_;
    const float* rbase = right + ((size_t)(b * C_) * H_ + h) * W_;

    const int tid = threadIdx.x;

    // ---- stage left: 32 cp x 64 w; two coalesced dword loads -> one b64 LDS store
#pragma unroll
    for (int it = 0; it < (CP * WCHUNK) / 128; ++it) {   // 16 iters
        int idx = it * 128 + tid;
        int cp  = idx >> 6;
        int w   = idx & 63;
        const float* p = lbase + (size_t)(2 * cp) * HW + w0 + w;
        v2f v;
        v.x = p[0];
        v.y = p[HW];          // folds into 24-bit instruction offset
        sL[cp * LS2 + w] = v;
    }
    // ---- stage right: 32 cp x 112 w'; w' = w0 - 48 + tid, clamped at left edge
    if (tid < RW) {
        int wg = w0 - 48 + tid;
        if (wg < 0) wg = 0;   // safe garbage; masked at store (w' < 0 => 0)
        const float* p = rbase + wg;
#pragma unroll
        for (int cp = 0; cp < CP; ++cp) {
            v2f v;
            v.x = p[0];
            v.y = p[HW];
            sR[cp * RS2 + tid] = v;
            p += 2 * (size_t)HW;
        }
    }
    __syncthreads();

    const int lane = tid & 31;
    const int wv   = tid >> 5;          // wave 0..3 -> 16-wide w tile
    const int hi   = (lane >> 4) & 1;   // lane half: K+2 (A/B), M+8 (D)
    const int lo16 = lane & 15;
    const int wloc = wv * 16;

    // Fragment base pointers; per k-step strides are immediate-foldable DS offsets.
    const v2f* pa = sL + hi * LS2 + wloc + lo16;   // + kk*2*LS2
    const v2f* pb = sR + hi * RS2 + wloc + lo16;   // + kk*2*RS2 + t*16

    // ---- hoist A fragments: a[kk] = A[M=lo16][K=4kk+2hi + {0,1}]
    v2f a[16];
#pragma unroll
    for (int kk = 0; kk < 16; ++kk)
        a[kk] = pa[kk * (2 * LS2)];

    v8f acc[4] = {};   // 4 column tiles: w' base = w0 + wloc - 48 + 16t
#pragma unroll
    for (int kk = 0; kk < 16; ++kk) {
#pragma unroll
        for (int t = 0; t < 4; ++t) {
            v2f bf = pb[kk * (2 * RS2) + t * 16];
            // D = A(16x4 f32) x B(4x16 f32) + C
            acc[t] = __builtin_amdgcn_wmma_f32_16x16x4_f32(
                false, a[kk], false, bf, (short)0, acc[t], false, false);
        }
    }

    // ---- masked scatter store: d = M - N + 48 - 16t; zero where w < d (w' < 0).
    // 4B scatters are assembled into full lines by the 192MB L2 before DRAM.
    float* obase = out + ((size_t)b * D_ * H_ + h) * W_;
#pragma unroll
    for (int t = 0; t < 4; ++t) {
#pragma unroll
        for (int r = 0; r < 8; ++r) {
            int M  = r + 8 * hi;
            int w  = w0 + wloc + M;
            int wp = w0 + wloc - 48 + 16 * t + lo16;   // w' (global)
            int d  = w - wp;
            if ((unsigned)d < (unsigned)D_) {
                float val = (wp >= 0) ? acc[t][r] * (1.0f / C_) : 0.0f;
                obase[(size_t)d * HW + w] = val;
            }
        }
    }
}

extern "C" void kernel_launch(void* const* d_in, const int* in_sizes, int n_in,
                              void* d_out, int out_size, void* d_ws, size_t ws_size,
                              hipStream_t stream) {
    const float* left  = (const float*)d_in[0];
    const float* right = (const float*)d_in[1];
    (void)in_sizes; (void)n_in; (void)out_size; (void)d_ws; (void)ws_size;
    float* out = (float*)d_out;

    dim3 grid(B_ * H_ * NWCHUNK);   // 6400 blocks
    dim3 block(128);                // 4 waves: one 16-w WMMA tile each
    hipLaunchKernelGGL(corr_volume_wmma, grid, block, 0, stream, left, right, out);
}